// RelationOnSpatial_3607772528788
// MI455X (gfx1250) — compile-verified
//
#include <hip/hip_runtime.h>
#include <hip/hip_bf16.h>

typedef float v2f __attribute__((ext_vector_type(2)));
typedef float v8f __attribute__((ext_vector_type(8)));

#define HW     16384   // 128*128 spatial positions
#define KDIM   2304    // 256 channels * 9 taps (dense)
#define KPAD   3072    // 256 channels * 12 taps (padded so 4 | taps-per-channel)
#define NROWS  1024
#define COUT   256

// K1: build padded W_eff: weffp[cy*12 + t] = sum_c w_p[c]*phi_w[c, cy*9+t] for t<9, 0 for t>=9.
//     Also biasphi = dot(w_p, phi_b).
__global__ void k_fold_phi(const float* __restrict__ phi_w, const float* __restrict__ phi_b,
                           const float* __restrict__ concat_w, float* __restrict__ weffp,
                           float* __restrict__ biasphi) {
    __shared__ float red[256];
    const float* wp = concat_w + 256;       // w_p = second half of concat weights
    if (blockIdx.x < 9) {
        int j = blockIdx.x * 256 + threadIdx.x;   // 0..2303 dense index
        float acc = 0.f;
        for (int c = 0; c < 256; ++c) acc += wp[c] * phi_w[c * KDIM + j];
        int cy = j / 9, t = j - cy * 9;
        weffp[cy * 12 + t] = acc;
    } else if (blockIdx.x == 9) {
        red[threadIdx.x] = wp[threadIdx.x] * phi_b[threadIdx.x];
        __syncthreads();
        for (int s = 128; s > 0; s >>= 1) {
            if (threadIdx.x < s) red[threadIdx.x] += red[threadIdx.x + s];
            __syncthreads();
        }
        if (threadIdx.x == 0) biasphi[0] = red[0];
    } else {
        // zero the pad taps t=9..11 (768 entries)
        for (int i = threadIdx.x; i < 768; i += 256) {
            int cy = i / 3, t = 9 + (i - cy * 3);
            weffp[cy * 12 + t] = 0.f;
        }
    }
}

// K2: s_phi = implicit-im2col(y) @ W_eff via V_WMMA_F32_16X16X4_F32.
// One wave per 16 consecutive spatial positions (same image row -> coalesced loads).
// Inner loop is div-free and branchless: per-lane tap offsets/masks precomputed,
// padded weights staged in LDS (B-fragment via LDS broadcast).
__global__ void k_sphi_wmma(const float* __restrict__ y, const float* __restrict__ weffp,
                            const float* __restrict__ biasphi, float* __restrict__ sphi) {
    __shared__ float wsh[KPAD];
    for (int i = threadIdx.x; i < KPAD; i += blockDim.x) wsh[i] = weffp[i];
    __syncthreads();

    int wave = (blockIdx.x * blockDim.x + threadIdx.x) >> 5;   // 0..1023 (M-tile)
    int lane = threadIdx.x & 31;
    int m    = lane & 15;
    int p    = wave * 16 + m;
    int h = p >> 7, w = p & 127;
    int ksel = (lane >> 4) << 1;      // lanes 0-15 -> taps +{0,1}; lanes 16-31 -> +{2,3}

    // Precompute per-lane y offsets and boundary masks for the 6 taps this lane touches
    // (3 chunks x 2 A-elements). Tap t in 0..11; t>=9 are zero-weight pads.
    int   offA[3][2];
    float mskA[3][2];
#pragma unroll
    for (int chunk = 0; chunk < 3; ++chunk) {
#pragma unroll
        for (int e = 0; e < 2; ++e) {
            int t  = chunk * 4 + ksel + e;
            int kh = t / 3, kw = t - kh * 3;
            int hh = h + kh - 1, ww = w + kw - 1;
            bool v = (t < 9) && hh >= 0 && hh < 128 && ww >= 0 && ww < 128;
            offA[chunk][e] = v ? (hh * 128 + ww) : 0;
            mskA[chunk][e] = v ? 1.f : 0.f;
        }
    }

    float bias = biasphi[0];
    v8f acc = {};
    for (int cy = 0; cy < 256; ++cy) {
        const float* yb = y + cy * HW;
        int wb = cy * 12 + ksel;
#pragma unroll
        for (int chunk = 0; chunk < 3; ++chunk) {
            v2f A = { yb[offA[chunk][0]] * mskA[chunk][0],
                      yb[offA[chunk][1]] * mskA[chunk][1] };
            v2f B = { wsh[wb + chunk * 4], wsh[wb + chunk * 4 + 1] };
            acc = __builtin_amdgcn_wmma_f32_16x16x4_f32(false, A, false, B, (short)0, acc,
                                                        false, false);
        }
    }
    // C/D layout: VGPR r, lane 0 -> (M=r, N=0); lane 16 -> (M=r+8, N=0)
    if (lane == 0) {
        for (int r = 0; r < 8; ++r) sphi[wave * 16 + r] = acc[r] + bias;
    } else if (lane == 16) {
        for (int r = 0; r < 8; ++r) sphi[wave * 16 + 8 + r] = acc[r] + bias;
    }
}

// K3: fvec = gamma * softmax(s_phi) over all 16384 entries (single block).
__global__ void k_softmax(const float* __restrict__ sphi, const float* __restrict__ gamma,
                          float* __restrict__ fvec) {
    __shared__ float red[1024];
    int t = threadIdx.x;
    float mx = -3.402823466e38f;
    for (int i = t; i < HW; i += 1024) mx = fmaxf(mx, sphi[i]);
    red[t] = mx; __syncthreads();
    for (int s = 512; s > 0; s >>= 1) {
        if (t < s) red[t] = fmaxf(red[t], red[t + s]);
        __syncthreads();
    }
    mx = red[0]; __syncthreads();
    float sum = 0.f;
    for (int i = t; i < HW; i += 1024) {
        float e = expf(sphi[i] - mx);
        fvec[i] = e;
        sum += e;
    }
    red[t] = sum; __syncthreads();
    for (int s = 512; s > 0; s >>= 1) {
        if (t < s) red[t] += red[t + s];
        __syncthreads();
    }
    float scale = gamma[0] / red[0];
    for (int i = t; i < HW; i += 1024) fvec[i] *= scale;
}

// K4: A[j] = sum_{h,w} fvec[h,w] * y_pad[cy, h+kh-1, w+kw-1], one block per j (2304)
__global__ void k_corr(const float* __restrict__ y, const float* __restrict__ fvec,
                       float* __restrict__ Avec) {
    __shared__ float red[256];
    int j = blockIdx.x;
    int cy = j / 9, r = j - cy * 9, kh = r / 3, kw = r - kh * 3;
    const float* yp = y + cy * HW + (kh - 1) * 128 + (kw - 1);
    float acc = 0.f;
    for (int p = threadIdx.x; p < HW; p += 256) {
        int h = p >> 7, w = p & 127;
        int hh = h + kh - 1, ww = w + kw - 1;
        float v = (hh >= 0 && hh < 128 && ww >= 0 && ww < 128) ? yp[p] : 0.f;
        acc += fvec[p] * v;
    }
    red[threadIdx.x] = acc; __syncthreads();
    for (int s = 128; s > 0; s >>= 1) {
        if (threadIdx.x < s) red[threadIdx.x] += red[threadIdx.x + s];
        __syncthreads();
    }
    if (threadIdx.x == 0) Avec[j] = red[0];
}

// K5: zbase[c] = sum_j g_w[c,j]*A[j] + gamma*g_b[c]  (256x2304 GEMV via WMMA, 16 waves).
// A_vec staged in LDS; B-fragment via LDS broadcast.
__global__ void k_zbase_wmma(const float* __restrict__ g_w, const float* __restrict__ g_b,
                             const float* __restrict__ gamma, const float* __restrict__ Avec,
                             float* __restrict__ zbase) {
    __shared__ float ash[KDIM];
    for (int i = threadIdx.x; i < KDIM; i += 512) ash[i] = Avec[i];
    __syncthreads();

    int wave = threadIdx.x >> 5;          // 0..15
    int lane = threadIdx.x & 31;
    int m    = lane & 15;
    const float* rowp = g_w + (wave * 16 + m) * (long)KDIM;
    int ksel = (lane >> 4) << 1;
    v8f acc = {};
    for (int k0 = 0; k0 < KDIM; k0 += 4) {
        int ka = k0 + ksel;
        v2f A = { rowp[ka], rowp[ka + 1] };
        v2f B = { ash[ka], ash[ka + 1] };
        acc = __builtin_amdgcn_wmma_f32_16x16x4_f32(false, A, false, B, (short)0, acc,
                                                    false, false);
    }
    float g = gamma[0];
    if (lane == 0) {
        for (int r = 0; r < 8; ++r) { int c = wave * 16 + r;     zbase[c] = acc[r] + g * g_b[c]; }
    } else if (lane == 16) {
        for (int r = 0; r < 8; ++r) { int c = wave * 16 + 8 + r; zbase[c] = acc[r] + g * g_b[c]; }
    }
}

// K6: out[n, c] = zbase[c] for all n (softmax over axis=1 makes every row identical)
__global__ void k_broadcast(const float* __restrict__ zbase, float* __restrict__ out) {
    __shared__ float zb[256];
    zb[threadIdx.x] = zbase[threadIdx.x];
    __syncthreads();
    out[(size_t)blockIdx.x * 256 + threadIdx.x] = zb[threadIdx.x];
}

extern "C" void kernel_launch(void* const* d_in, const int* in_sizes, int n_in,
                              void* d_out, int out_size, void* d_ws, size_t ws_size,
                              hipStream_t stream) {
    (void)in_sizes; (void)n_in; (void)out_size; (void)ws_size;
    // setup_inputs order: x, y, g_w, g_b, phi_w, phi_b, theta_w, theta_b, concat_w, gamma
    const float* y        = (const float*)d_in[1];
    const float* g_w      = (const float*)d_in[2];
    const float* g_b      = (const float*)d_in[3];
    const float* phi_w    = (const float*)d_in[4];
    const float* phi_b    = (const float*)d_in[5];
    const float* concat_w = (const float*)d_in[8];
    const float* gamma    = (const float*)d_in[9];
    float* out = (float*)d_out;

    float* ws      = (float*)d_ws;
    float* weffp   = ws;                    // 3072 (padded 256x12)
    float* biasphi = ws + KPAD;             // 1 (+63 pad)
    float* sphi    = ws + KPAD + 64;        // 16384
    float* fvec    = sphi + HW;             // 16384
    float* Avec    = fvec + HW;             // 2304
    float* zbase   = Avec + KDIM;           // 256

    k_fold_phi  <<<11,   256, 0, stream>>>(phi_w, phi_b, concat_w, weffp, biasphi);
    k_sphi_wmma <<<128,  256, 0, stream>>>(y, weffp, biasphi, sphi);   // 1024 waves
    k_softmax   <<<1,   1024, 0, stream>>>(sphi, gamma, fvec);
    k_corr      <<<KDIM, 256, 0, stream>>>(y, fvec, Avec);
    k_zbase_wmma<<<1,    512, 0, stream>>>(g_w, g_b, gamma, Avec, zbase);
    k_broadcast <<<NROWS,256, 0, stream>>>(zbase, out);
}